// GAT_47004122087945
// MI455X (gfx1250) — compile-verified
//
#include <hip/hip_runtime.h>

// ---------------------------------------------------------------------------
// 3-layer GAT for MI455X (gfx1250). GEMMs use v_wmma_f32_16x16x4_f32 (full
// fp32 precision, matching the reference). Edge softmax/scatter uses
// hardware f32 atomics. All scratch in d_ws; deterministic per call.
// ---------------------------------------------------------------------------

typedef float v2f __attribute__((ext_vector_type(2)));
typedef float v8f __attribute__((ext_vector_type(8)));

namespace {
constexpr int N_NODES = 100000;
constexpr int N_EDGES = 1600000;
constexpr int E_TOT   = N_EDGES + N_NODES;   // with self loops
constexpr int IN_C    = 64;
constexpr int HID_C   = 32;
constexpr int OUT_C   = 16;
constexpr int HEADS   = 4;
constexpr int F_HID   = HEADS * HID_C;       // 128
}

// order-preserving encode of f32 into u32 for atomicMax-based segment max
__device__ __forceinline__ unsigned enc_ord(float x) {
  unsigned u = __float_as_uint(x);
  return (u & 0x80000000u) ? ~u : (u | 0x80000000u);
}
__device__ __forceinline__ float dec_ord(unsigned k) {
  unsigned u = (k & 0x80000000u) ? (k & 0x7fffffffu) : ~k;
  return __uint_as_float(u);
}

// ---------------------------------------------------------------------------
// zero a 32-bit buffer
// ---------------------------------------------------------------------------
__global__ void zero32_kernel(unsigned* __restrict__ p, long long n) {
  long long t = (long long)blockIdx.x * blockDim.x + threadIdx.x;
  if (t < n) p[t] = 0u;
}

// ---------------------------------------------------------------------------
// WMMA GEMM: Hout[N,F] = X[N,K] * W[K,F], fp32, one wave per 16x16 tile.
// Lane layouts per CDNA5 ISA 7.12.2:
//   A 16x4 : lane l(0..15) v0=A[l][k0], v1=A[l][k0+1]; upper half k0+=2
//   B 4x16 : lane l(0..15) v0=B[k0][l], v1=B[k0+1][l]; upper half k0+=2
//   C 16x16: vgpr r, lanes 0-15 -> M=r, lanes 16-31 -> M=r+8
// ---------------------------------------------------------------------------
__global__ void wmma_gemm_kernel(const float* __restrict__ X,
                                 const float* __restrict__ W,
                                 float* __restrict__ Hout,
                                 int N, int K, int F) {
  int wave = (int)(((long long)blockIdx.x * blockDim.x + threadIdx.x) >> 5);
  int lane = threadIdx.x & 31;
  int tilesF = F >> 4;
  int nTiles = (N >> 4) * tilesF;
  if (wave >= nTiles) return;                 // wave-uniform guard
  int m0 = (wave / tilesF) << 4;
  int n0 = (wave % tilesF) << 4;
  int half = lane >> 4;                       // 0 or 1
  int l    = lane & 15;

  const float* xp = X + (size_t)(m0 + l) * K + (half << 1);
  const float* wp = W + (size_t)(half << 1) * F + n0 + l;

  v8f acc = {};
  for (int k = 0; k < K; k += 4) {
    v2f a, b;
    a.x = xp[k];
    a.y = xp[k + 1];
    b.x = wp[(size_t)k * F];
    b.y = wp[(size_t)(k + 1) * F];
    acc = __builtin_amdgcn_wmma_f32_16x16x4_f32(
        /*neg_a=*/false, a, /*neg_b=*/false, b,
        /*c_mod=*/(short)0, acc, /*reuse_a=*/false, /*reuse_b=*/false);
  }

  float* op = Hout + (size_t)(m0 + (half << 3)) * F + n0 + l;
#pragma unroll
  for (int r = 0; r < 8; ++r) op[(size_t)r * F] = acc[r];
}

// ---------------------------------------------------------------------------
// per-(node,head) attention score dots: s_dst = h . att[:C], s_src = h . att[C:]
// ---------------------------------------------------------------------------
__global__ void score_kernel(const float* __restrict__ h,
                             const float* __restrict__ att,
                             float* __restrict__ s_dst, float* __restrict__ s_src,
                             int H, int C) {
  int t = blockIdx.x * blockDim.x + threadIdx.x;
  if (t >= N_NODES * H) return;
  int n = t / H, hd = t % H;
  const float* hp = h + (size_t)n * H * C + (size_t)hd * C;
  const float* ad = att + (size_t)hd * 2 * C;
  const float* as = ad + C;
  float sd = 0.f, ss = 0.f;
  for (int c = 0; c < C; ++c) {
    float v = hp[c];
    sd += v * ad[c];
    ss += v * as[c];
  }
  s_dst[t] = sd;
  s_src[t] = ss;
}

// ---------------------------------------------------------------------------
// materialize int32 src/dst with self loops (edge_index is int64 [2,E])
// ---------------------------------------------------------------------------
__global__ void prep_edges_kernel(const long long* __restrict__ ei,
                                  int* __restrict__ srcI, int* __restrict__ dstI) {
  int t = blockIdx.x * blockDim.x + threadIdx.x;
  if (t >= E_TOT) return;
  if (t < N_EDGES) {
    srcI[t] = (int)ei[t];
    dstI[t] = (int)ei[N_EDGES + t];
  } else {
    srcI[t] = t - N_EDGES;
    dstI[t] = t - N_EDGES;
  }
}

// ---------------------------------------------------------------------------
// edge pass A: e = leaky_relu(s_dst[dst]+s_src[src], 0.2); segment max via
// order-encoded atomicMax
// ---------------------------------------------------------------------------
__global__ void edge_score_kernel(const int* __restrict__ srcI,
                                  const int* __restrict__ dstI,
                                  const float* __restrict__ s_dst,
                                  const float* __restrict__ s_src,
                                  float* __restrict__ ebuf,
                                  unsigned* __restrict__ menc, int H) {
  long long t = (long long)blockIdx.x * blockDim.x + threadIdx.x;
  if (t >= (long long)E_TOT * H) return;
  int edge = (int)(t / H), hd = (int)(t % H);
  int s = srcI[edge], d = dstI[edge];
  float x = s_dst[d * H + hd] + s_src[s * H + hd];
  x = x > 0.f ? x : 0.2f * x;
  ebuf[t] = x;
  atomicMax(&menc[d * H + hd], enc_ord(x));
}

// ---------------------------------------------------------------------------
// edge pass B: a = exp(e - m[dst]); acc[dst] += a*h[src]; denom[dst] += a
// (normalization deferred to node epilogue)
// ---------------------------------------------------------------------------
__global__ void edge_scatter_kernel(const int* __restrict__ srcI,
                                    const int* __restrict__ dstI,
                                    const float* __restrict__ h,
                                    const float* __restrict__ ebuf,
                                    const unsigned* __restrict__ menc,
                                    float* __restrict__ acc,
                                    float* __restrict__ denom,
                                    int H, int C) {
  const int F = H * C;
  long long t = (long long)blockIdx.x * blockDim.x + threadIdx.x;
  if (t >= (long long)E_TOT * F) return;
  int edge = (int)(t / F), f = (int)(t % F);
  int hd = f / C;
  int s = srcI[edge], d = dstI[edge];
  float m = dec_ord(menc[d * H + hd]);
  float a = __expf(ebuf[(long long)edge * H + hd] - m);
  unsafeAtomicAdd(&acc[(size_t)d * F + f], a * h[(size_t)s * F + f]);
  if (f == hd * C) unsafeAtomicAdd(&denom[d * H + hd], a);
}

// ---------------------------------------------------------------------------
// node epilogues
// ---------------------------------------------------------------------------
__global__ void finalize_elu_kernel(const float* __restrict__ acc,
                                    const float* __restrict__ denom,
                                    float* __restrict__ act, int H, int C) {
  const int F = H * C;
  long long t = (long long)blockIdx.x * blockDim.x + threadIdx.x;
  if (t >= (long long)N_NODES * F) return;
  int n = (int)(t / F), f = (int)(t % F);
  float v = acc[t] / denom[n * H + f / C];
  act[t] = v > 0.f ? v : (__expf(v) - 1.0f);   // ELU, alpha=1
}

__global__ void finalize_out_kernel(const float* __restrict__ acc,
                                    const float* __restrict__ denom,
                                    float* __restrict__ out) {
  // layer 3: H=1, C=OUT_C; mean over 1 head == identity
  long long t = (long long)blockIdx.x * blockDim.x + threadIdx.x;
  if (t >= (long long)N_NODES * OUT_C) return;
  int n = (int)(t / OUT_C);
  out[t] = acc[t] / denom[n];
}

// ---------------------------------------------------------------------------
// launcher
// ---------------------------------------------------------------------------
static inline unsigned blk(long long n, int t) { return (unsigned)((n + t - 1) / t); }

extern "C" void kernel_launch(void* const* d_in, const int* in_sizes, int n_in,
                              void* d_out, int out_size, void* d_ws, size_t ws_size,
                              hipStream_t stream) {
  (void)in_sizes; (void)n_in; (void)out_size; (void)ws_size;

  const float*     x    = (const float*)d_in[0];
  const long long* ei   = (const long long*)d_in[1];   // int64 [2, N_EDGES]
  const float*     W1   = (const float*)d_in[2];
  const float*     att1 = (const float*)d_in[3];
  const float*     W2   = (const float*)d_in[4];
  const float*     att2 = (const float*)d_in[5];
  const float*     W3   = (const float*)d_in[6];
  const float*     att3 = (const float*)d_in[7];
  float* out = (float*)d_out;

  // workspace carve-up (32-bit words)
  float*    A    = (float*)d_ws;                            // gemm out h : N*128
  float*    B    = A  + (size_t)N_NODES * F_HID;            // acc        : N*128
  float*    Cb   = B  + (size_t)N_NODES * F_HID;            // activation : N*128
  float*    sd   = Cb + (size_t)N_NODES * F_HID;            // s_dst      : N*H
  float*    ss   = sd + (size_t)N_NODES * HEADS;            // s_src      : N*H
  unsigned* me   = (unsigned*)(ss + (size_t)N_NODES * HEADS); // seg max  : N*H
  float*    dn   = (float*)(me + (size_t)N_NODES * HEADS);  // denom      : N*H
  float*    ebuf = dn + (size_t)N_NODES * HEADS;            // edge e     : E_TOT*H
  int*      srcI = (int*)(ebuf + (size_t)E_TOT * HEADS);    // src idx    : E_TOT
  int*      dstI = srcI + (size_t)E_TOT;                    // dst idx    : E_TOT

  const int T = 256;

  prep_edges_kernel<<<blk(E_TOT, T), T, 0, stream>>>(ei, srcI, dstI);

  auto run_layer = [&](const float* Xin, int K, const float* W, const float* att,
                       int H, int C, bool last) {
    const int F = H * C;
    // GEMM: one wave (32 threads) per 16x16 tile
    long long tiles = (long long)(N_NODES / 16) * (F / 16);
    wmma_gemm_kernel<<<blk(tiles * 32, T), T, 0, stream>>>(Xin, W, A, N_NODES, K, F);
    // scores
    score_kernel<<<blk((long long)N_NODES * H, T), T, 0, stream>>>(A, att, sd, ss, H, C);
    // clear accumulators for this layer
    zero32_kernel<<<blk((long long)N_NODES * F, T), T, 0, stream>>>((unsigned*)B,
                                                                    (long long)N_NODES * F);
    zero32_kernel<<<blk((long long)N_NODES * H, T), T, 0, stream>>>(me,
                                                                    (long long)N_NODES * H);
    zero32_kernel<<<blk((long long)N_NODES * H, T), T, 0, stream>>>((unsigned*)dn,
                                                                    (long long)N_NODES * H);
    // edge passes
    edge_score_kernel<<<blk((long long)E_TOT * H, T), T, 0, stream>>>(srcI, dstI, sd, ss,
                                                                      ebuf, me, H);
    edge_scatter_kernel<<<blk((long long)E_TOT * F, T), T, 0, stream>>>(srcI, dstI, A, ebuf,
                                                                        me, B, dn, H, C);
    // node epilogue
    if (!last) {
      finalize_elu_kernel<<<blk((long long)N_NODES * F, T), T, 0, stream>>>(B, dn, Cb, H, C);
    } else {
      finalize_out_kernel<<<blk((long long)N_NODES * OUT_C, T), T, 0, stream>>>(B, dn, out);
    }
  };

  run_layer(x,  IN_C,  W1, att1, HEADS, HID_C, false);   // layer 1: [N,64]->[N,128]
  run_layer(Cb, F_HID, W2, att2, HEADS, HID_C, false);   // layer 2: [N,128]->[N,128]
  run_layer(Cb, F_HID, W3, att3, 1,     OUT_C, true);    // layer 3: [N,128]->[N,16]
}